// InterSTELLAR_28166395527246
// MI455X (gfx1250) — compile-verified
//
#include <hip/hip_runtime.h>
#include <math.h>

// ---------------------------------------------------------------------------
// InterSTELLAR GCN pipeline for MI455X (gfx1250, wave32, WMMA + TDM).
// Dense projections use V_WMMA_F32_16X16X4_F32 (exact f32). GEMM X-tiles are
// staged into LDS by the Tensor Data Mover (1-D contiguous descriptor) when
// K needs no LDS padding, else by a zero-padded manual path. The workload is
// bound by ~1.3 GB of edge gather/scatter traffic (~60us at 23.3 TB/s).
// ---------------------------------------------------------------------------

#define IN_C 30
#define HID  40
#define D1   20
#define D2   10
#define NCLS 3

typedef __attribute__((ext_vector_type(2))) float v2f;
typedef __attribute__((ext_vector_type(8))) float v8f;
typedef __attribute__((ext_vector_type(4))) unsigned int uint4v;
typedef __attribute__((ext_vector_type(8))) int int8v;
typedef __attribute__((ext_vector_type(4))) int int4v;

__device__ __forceinline__ float selu_f(float x) {
    const float a = 1.6732632423543772f;
    const float s = 1.0507009873554805f;
    return s * (x > 0.f ? x : a * (__expf(x) - 1.f));
}

// ---------------------------------------------------------------------------
// TDM: 1-D contiguous copy of nElem dwords, global -> LDS.
// Descriptor packing per CDNA5 ISA:
//   group0: [1:0]=count=1 | [63:32]=lds_addr | [120:64]=global_addr | type=2
//   group1: data_size=4B(2); tensor_dim0=nElem; tensor_dim1=1; tile_dim0=nElem;
//           tile_dim1/2 unused; tensor_dim0_stride=nElem; no pad/iterate/mask.
//   groups 2/3 (+extension): zero (dims 2..4 unused).
// This toolchain exposes the 6-arg builtin:
//   (uint32x4, int32x8, int32x4, int32x4, int32x8, i32 cpol)
// ---------------------------------------------------------------------------
__device__ __forceinline__ void tdm_load_1d(unsigned int lds_off,
                                            const void* gptr,
                                            unsigned int nElem)
{
    unsigned long long ga = (unsigned long long)(uintptr_t)gptr;
    uint4v g0;
    g0.x = 1u;                                                  // count=1
    g0.y = lds_off;                                             // LDS byte addr
    g0.z = (unsigned int)ga;                                    // addr[31:0]
    g0.w = (unsigned int)((ga >> 32) & 0x01FFFFFFu) | (2u << 30); // addr[56:32]|type=2
    int8v g1;
    g1[0] = 0x00020000;                        // data_size=2 (4 bytes)
    g1[1] = (int)((nElem & 0xFFFFu) << 16);    // tensor_dim0[15:0]  (bits 63:48)
    g1[2] = (int)((nElem >> 16) & 0xFFFFu) | (1 << 16); // tensor_dim0[31:16], tensor_dim1=1
    g1[3] = (int)((nElem & 0xFFFFu) << 16);    // tile_dim0 (bits 127:112)
    g1[4] = 0;                                 // tile_dim1/tile_dim2 unused
    g1[5] = (int)nElem;                        // tensor_dim0_stride[31:0]
    g1[6] = 0;
    g1[7] = 0;
    int4v z4 = {0, 0, 0, 0};
    int8v z8 = {0, 0, 0, 0, 0, 0, 0, 0};
    __builtin_amdgcn_tensor_load_to_lds(g0, g1, z4, z4, z8, 0);
}

// ---------------------------------------------------------------------------
// f32 WMMA GEMM:  Y[M x F] = X[M x K] @ W[K x F]   (row-major, K/F static)
// Block = 256 threads = 8 waves; block computes 128 rows x 16 cols.
// LDS: sX[128][KP] X tile; sW[16][KP] = W^T slice (zero-padded).
// Full tiles with KP==K are staged by the TDM (one DMA per block, wave 0
// issues it, s_wait_tensorcnt + barrier publishes); otherwise manual staging.
// MODE 0: plain store.  MODE 1: Y = selu(Y + bias[col]).
// ---------------------------------------------------------------------------
template <int K, int F, int MODE>
__global__ void wmma_gemm_f32(const float* __restrict__ X,
                              const float* __restrict__ W,
                              const float* __restrict__ bias,
                              float* __restrict__ Y, int M)
{
    constexpr int KP = (K + 3) & ~3;            // K padded to multiple of 4
    __shared__ __align__(16) float sX[128 * KP];
    __shared__ __align__(16) float sW[16 * KP];

    const int tid   = threadIdx.x;
    const int rows0 = blockIdx.x * 128;
    const int cols0 = blockIdx.y * 16;

    constexpr bool tdmOK = (KP == K);           // dense LDS layout == global layout
    const bool fullTile = tdmOK && (rows0 + 128 <= M);

    if (fullTile) {
        if (tid < 32) {                         // wave 0 issues the DMA
            tdm_load_1d((unsigned int)(uintptr_t)(void*)sX,
                        X + (long)rows0 * K, 128u * K);
            __builtin_amdgcn_s_wait_tensorcnt(0);
        }
    } else {
        // manual zero-padded staging (K tail / row tail)
        for (int i = tid; i < 128 * KP; i += 256) {
            int r = i / KP, k = i - r * KP;
            int row = rows0 + r;
            sX[i] = (row < M && k < K) ? X[(long)row * K + k] : 0.f;
        }
    }
    // stage W^T slice: sW[c][k] = W[k][cols0+c]
    for (int i = tid; i < 16 * KP; i += 256) {
        int c = i / KP, k = i - c * KP;
        int col = cols0 + c;
        sW[i] = (col < F && k < K) ? W[(long)k * F + col] : 0.f;
    }
    __syncthreads();

    const int lane = tid & 31;
    const int wave = tid >> 5;
    const int half = lane >> 4;                 // 0 -> K pair {0,1}; 1 -> {2,3}
    const int l    = lane & 15;

    const float* aRow = &sX[(wave * 16 + l) * KP];
    const float* bCol = &sW[l * KP];

    v8f c = {0.f, 0.f, 0.f, 0.f, 0.f, 0.f, 0.f, 0.f};
#pragma unroll
    for (int ks = 0; ks < KP / 4; ++ks) {
        const int kb = ks * 4 + half * 2;       // even -> 8B-aligned ds_load_b64
        v2f a = *(const v2f*)(aRow + kb);
        v2f b = *(const v2f*)(bCol + kb);
        c = __builtin_amdgcn_wmma_f32_16x16x4_f32(
                false, a, false, b, (short)0, c, false, false);
    }

    const int col     = cols0 + l;
    const int rowBase = rows0 + wave * 16 + half * 8;
    if (col < F) {
        float bcol = (MODE == 1) ? bias[col] : 0.f;
        float* yp = Y + (long)rowBase * F + col;
        if (rowBase + 8 <= M) {                 // fast path: 8 stores, one mask
#pragma unroll
            for (int r = 0; r < 8; ++r) {
                float v = c[r];
                if (MODE == 1) v = selu_f(v + bcol);
                yp[(long)r * F] = v;
            }
        } else {                                // tail rows
#pragma unroll
            for (int r = 0; r < 8; ++r) {
                if (rowBase + r < M) {
                    float v = c[r];
                    if (MODE == 1) v = selu_f(v + bcol);
                    yp[(long)r * F] = v;
                }
            }
        }
    }
}

// ---------------------------------------------------------------------------
// Small utility kernels
// ---------------------------------------------------------------------------
__global__ void zero_f(float* __restrict__ p, long n) {
    long i = (long)blockIdx.x * blockDim.x + threadIdx.x;
    if (i < n) p[i] = 0.f;
}

__global__ void deg_init(float* __restrict__ deg, int N) {
    int n = blockIdx.x * blockDim.x + threadIdx.x;
    if (n < N) deg[n] = 1.0f;   // +1 self-loop weight
}

__global__ void deg_acc(const long long* __restrict__ dst,
                        const float* __restrict__ ew,
                        float* __restrict__ deg, int E) {
    int e = blockIdx.x * blockDim.x + threadIdx.x;
    if (e < E) atomicAdd(&deg[(int)dst[e]], ew[e]);
}

__global__ void deg_rsqrt(float* __restrict__ deg, int N) {
    int n = blockIdx.x * blockDim.x + threadIdx.x;
    if (n < N) deg[n] = rsqrtf(deg[n]);   // deg >= 1 always
}

// agg[dst] += h[src] * (dinv[src]*ew*dinv[dst]), 40 features per edge
__global__ void edge_scatter(const long long* __restrict__ src,
                             const long long* __restrict__ dst,
                             const float* __restrict__ ew,
                             const float* __restrict__ dinv,
                             const float* __restrict__ h,
                             float* __restrict__ agg, int E)
{
    int e = blockIdx.x * blockDim.x + threadIdx.x;
    if (e >= E) return;
    int s = (int)src[e], d = (int)dst[e];
    const float* hr = h + (long)s * HID;
    __builtin_prefetch(hr, 0, 1);               // global_prefetch_b8: start the
                                                // random gather early
    float nrm = dinv[s] * ew[e] * dinv[d];
    float* ar = agg + (long)d * HID;
#pragma unroll 8
    for (int f = 0; f < HID; ++f) atomicAdd(&ar[f], hr[f] * nrm);
}

// out = selu(layernorm(agg + h*dinv^2 + b))  per node, HID features
__global__ void gcn_post(const float* __restrict__ agg,
                         const float* __restrict__ h,
                         const float* __restrict__ dinv,
                         const float* __restrict__ b,
                         const float* __restrict__ lnw,
                         const float* __restrict__ lnb,
                         float* __restrict__ out, int N)
{
    int n = blockIdx.x * blockDim.x + threadIdx.x;
    if (n >= N) return;
    float dv = dinv[n];
    float d2 = dv * dv;
    const float* ar = agg + (long)n * HID;
    const float* hr = h + (long)n * HID;
    float v[HID];
    float mean = 0.f;
#pragma unroll
    for (int f = 0; f < HID; ++f) {
        float x = ar[f] + hr[f] * d2 + b[f];
        v[f] = x; mean += x;
    }
    mean *= (1.f / HID);
    float var = 0.f;
#pragma unroll
    for (int f = 0; f < HID; ++f) { float d = v[f] - mean; var += d * d; }
    var *= (1.f / HID);
    float inv = rsqrtf(var + 1e-5f);
    float* o = out + (long)n * HID;
#pragma unroll
    for (int f = 0; f < HID; ++f)
        o[f] = selu_f((v[f] - mean) * inv * lnw[f] + lnb[f]);
}

// score[n] = (tanh(x3@Wt+bt) * sigmoid(x3@Ws+bs)) @ Wc + bc
__global__ void attn_scores(const float* __restrict__ x3,
                            const float* __restrict__ Wt, const float* __restrict__ bt,
                            const float* __restrict__ Ws, const float* __restrict__ bs,
                            const float* __restrict__ Wc, const float* __restrict__ bc,
                            float* __restrict__ score, int N)
{
    int n = blockIdx.x * blockDim.x + threadIdx.x;
    if (n >= N) return;
    const float* r = x3 + (long)n * D1;
    float rl[D1];
#pragma unroll
    for (int i = 0; i < D1; ++i) rl[i] = r[i];
    float sc = bc[0];
#pragma unroll
    for (int j = 0; j < D2; ++j) {
        float t = bt[j], sg = bs[j];
#pragma unroll
        for (int i = 0; i < D1; ++i) {
            t  += rl[i] * Wt[i * D2 + j];
            sg += rl[i] * Ws[i * D2 + j];
        }
        float g = tanhf(t) * (1.f / (1.f + __expf(-sg)));
        sc += g * Wc[j];
    }
    score[n] = sc;
}

// single block: misc[0] = max(score), misc[1] = sum(exp(score - max))
__global__ void softmax_reduce(const float* __restrict__ s, float* __restrict__ misc, int N)
{
    __shared__ float red[1024];
    int tid = threadIdx.x;
    float m = -3.4e38f;
    for (int n = tid; n < N; n += 1024) m = fmaxf(m, s[n]);
    red[tid] = m; __syncthreads();
    for (int k = 512; k > 0; k >>= 1) {
        if (tid < k) red[tid] = fmaxf(red[tid], red[tid + k]);
        __syncthreads();
    }
    m = red[0]; __syncthreads();
    float sum = 0.f;
    for (int n = tid; n < N; n += 1024) sum += __expf(s[n] - m);
    red[tid] = sum; __syncthreads();
    for (int k = 512; k > 0; k >>= 1) {
        if (tid < k) red[tid] += red[tid + k];
        __syncthreads();
    }
    if (tid == 0) { misc[0] = m; misc[1] = red[0]; }
}

// A[n] = exp(s[n]-m)/Z; pooled += A[n]*x3[n]  (LDS partials -> global atomics)
__global__ void softmax_apply_pool(const float* __restrict__ score,
                                   const float* __restrict__ misc,
                                   const float* __restrict__ x3,
                                   float* __restrict__ A,
                                   float* __restrict__ pooled, int N)
{
    __shared__ float part[D1];
    int tid = threadIdx.x;
    if (tid < D1) part[tid] = 0.f;
    __syncthreads();
    int n = blockIdx.x * blockDim.x + tid;
    if (n < N) {
        float a = __expf(score[n] - misc[0]) / misc[1];
        A[n] = a;
        const float* r = x3 + (long)n * D1;
#pragma unroll
        for (int f = 0; f < D1; ++f) atomicAdd(&part[f], a * r[f]);
    }
    __syncthreads();
    if (tid < D1) atomicAdd(&pooled[tid], part[tid]);
}

// iterative top-8 (by A) and bottom-8 (top-8 of -A), first-index tiebreak
__global__ void topk_kernel(const float* __restrict__ A, int* __restrict__ idx_out, int N)
{
    __shared__ float bv[1024];
    __shared__ int   bi[1024];
    __shared__ int   chosen[16];
    const int tid = threadIdx.x;
    for (int phase = 0; phase < 2; ++phase) {
        const float sgn = phase ? -1.f : 1.f;
        for (int t = 0; t < 8; ++t) {
            float best = -3.4e38f; int besti = 0x7fffffff;
            for (int n = tid; n < N; n += 1024) {
                bool skip = false;
                for (int j = 0; j < t; ++j)
                    if (chosen[phase * 8 + j] == n) skip = true;
                if (skip) continue;
                float v = sgn * A[n];
                if (v > best || (v == best && n < besti)) { best = v; besti = n; }
            }
            bv[tid] = best; bi[tid] = besti;
            __syncthreads();
            for (int k = 512; k > 0; k >>= 1) {
                if (tid < k) {
                    if (bv[tid + k] > bv[tid] ||
                        (bv[tid + k] == bv[tid] && bi[tid + k] < bi[tid])) {
                        bv[tid] = bv[tid + k]; bi[tid] = bi[tid + k];
                    }
                }
                __syncthreads();
            }
            if (tid == 0) { chosen[phase * 8 + t] = bi[0]; idx_out[phase * 8 + t] = bi[0]; }
            __syncthreads();
        }
    }
}

// 1 thread: bag logits/softmax/argmax + SmoothTop1SVM cell loss
__global__ void finalize_kernel(const float* __restrict__ pooled,
                                const float* __restrict__ Wcls, const float* __restrict__ bcls,
                                const float* __restrict__ x3,
                                const float* __restrict__ Wcell, const float* __restrict__ bcell,
                                const int* __restrict__ labelp,
                                const int* __restrict__ idx,
                                float* __restrict__ out, int N)
{
    if (threadIdx.x != 0 || blockIdx.x != 0) return;
    const int label = labelp[0];
    float logits[NCLS];
    for (int c = 0; c < NCLS; ++c) {
        float s = bcls[c];
        for (int i = 0; i < D1; ++i) s += pooled[i] * Wcls[i * NCLS + c];
        logits[c] = s;
    }
    float mx = logits[0]; int am = 0;
    for (int c = 1; c < NCLS; ++c) if (logits[c] > mx) { mx = logits[c]; am = c; }
    float Z = 0.f;
    for (int c = 0; c < NCLS; ++c) Z += __expf(logits[c] - mx);
    for (int c = 0; c < NCLS; ++c) out[c] = logits[c];
    for (int c = 0; c < NCLS; ++c) out[NCLS + c] = __expf(logits[c] - mx) / Z;
    out[2 * NCLS] = (float)am;

    const float* Wl = Wcell + (long)label * D1 * 2;
    const float* bl = bcell + (long)label * 2;
    float loss = 0.f;
    for (int c = 0; c < 16; ++c) {
        int n = idx[c];
        int target = (c < 8) ? 1 : 0;
        float l0 = bl[0], l1 = bl[1];
        const float* r = x3 + (long)n * D1;
        for (int i = 0; i < D1; ++i) {
            l0 += r[i] * Wl[i * 2 + 0];
            l1 += r[i] * Wl[i * 2 + 1];
        }
        float d0 = (target == 0) ? 0.f : 1.f;  // delta = 1 - one_hot(target)
        float d1 = (target == 1) ? 0.f : 1.f;
        float a0 = l0 + d0, a1 = l1 + d1;
        float m = fmaxf(a0, a1);
        float lse = m + __logf(__expf(a0 - m) + __expf(a1 - m));
        float sy = target ? l1 : l0;
        loss += lse - sy;
    }
    out[7 + 21L * N] = loss * (1.f / 16.f);
}

// ---------------------------------------------------------------------------
// Host-side launch
// ---------------------------------------------------------------------------
extern "C" void kernel_launch(void* const* d_in, const int* in_sizes, int n_in,
                              void* d_out, int out_size, void* d_ws, size_t ws_size,
                              hipStream_t stream)
{
    (void)n_in; (void)out_size; (void)ws_size;
    const float*     x     = (const float*)d_in[0];
    const long long* ei    = (const long long*)d_in[1];
    const float*     ew    = (const float*)d_in[2];
    const int*       label = (const int*)d_in[3];
    const float* W1  = (const float*)d_in[4];
    const float* b1  = (const float*)d_in[5];
    const float* l1w = (const float*)d_in[6];
    const float* l1b = (const float*)d_in[7];
    const float* W2  = (const float*)d_in[8];
    const float* b2  = (const float*)d_in[9];
    const float* l2w = (const float*)d_in[10];
    const float* l2b = (const float*)d_in[11];
    const float* Wf  = (const float*)d_in[12];
    const float* bf  = (const float*)d_in[13];
    const float* Wt  = (const float*)d_in[14];
    const float* bt  = (const float*)d_in[15];
    const float* Ws  = (const float*)d_in[16];
    const float* bs  = (const float*)d_in[17];
    const float* Wc  = (const float*)d_in[18];
    const float* bc  = (const float*)d_in[19];
    const float* Wcls  = (const float*)d_in[20];
    const float* bcls  = (const float*)d_in[21];
    const float* Wcell = (const float*)d_in[22];
    const float* bcell = (const float*)d_in[23];

    float* out = (float*)d_out;
    float* ws  = (float*)d_ws;

    const int N = in_sizes[0] / IN_C;   // 100000
    const int E = in_sizes[2];          // 2000000
    const long long* src = ei;
    const long long* dst = ei + E;

    // workspace layout (floats): deg/dinv | score | bufA(N*40) | bufB(N*40) | misc(128)
    float* deg    = ws;
    float* score  = ws + N;
    float* bufA   = ws + 2L * N;
    float* bufB   = ws + 2L * N + (long)HID * N;
    float* misc   = ws + 2L * N + 2L * HID * N;
    float* pooled = misc + 2;
    int*   idx    = (int*)(misc + 32);

    float* A_out  = out + 7;
    float* x3_out = out + 7 + N;

    const int nbN  = (N + 255) / 256;
    const int nbE  = (E + 255) / 256;
    const int nbNF = (N * HID + 255) / 256;
    const int gx   = (N + 127) / 128;           // 128 rows per block (8 waves)

    deg_init<<<nbN, 256, 0, stream>>>(deg, N);
    zero_f<<<1, 128, 0, stream>>>(misc, 128);
    deg_acc<<<nbE, 256, 0, stream>>>(dst, ew, deg, E);
    deg_rsqrt<<<nbN, 256, 0, stream>>>(deg, N);

    // --- GCN layer 1 ---
    wmma_gemm_f32<IN_C, HID, 0><<<dim3(gx, 3), 256, 0, stream>>>(x, W1, nullptr, bufA, N);
    zero_f<<<nbNF, 256, 0, stream>>>(bufB, (long)N * HID);
    edge_scatter<<<nbE, 256, 0, stream>>>(src, dst, ew, deg, bufA, bufB, E);
    gcn_post<<<nbN, 256, 0, stream>>>(bufB, bufA, deg, b1, l1w, l1b, bufB, N);

    // --- GCN layer 2 ---
    wmma_gemm_f32<HID, HID, 0><<<dim3(gx, 3), 256, 0, stream>>>(bufB, W2, nullptr, bufA, N);
    zero_f<<<nbNF, 256, 0, stream>>>(bufB, (long)N * HID);
    edge_scatter<<<nbE, 256, 0, stream>>>(src, dst, ew, deg, bufA, bufB, E);
    gcn_post<<<nbN, 256, 0, stream>>>(bufB, bufA, deg, b2, l2w, l2b, bufB, N);

    // --- FFN: x3 = selu(x2 @ Wf + bf), written straight into d_out ---
    wmma_gemm_f32<HID, D1, 1><<<dim3(gx, 2), 256, 0, stream>>>(bufB, Wf, bf, x3_out, N);

    // --- attention + softmax + pooling ---
    attn_scores<<<nbN, 256, 0, stream>>>(x3_out, Wt, bt, Ws, bs, Wc, bc, score, N);
    softmax_reduce<<<1, 1024, 0, stream>>>(score, misc, N);
    softmax_apply_pool<<<nbN, 256, 0, stream>>>(score, misc, x3_out, A_out, pooled, N);

    // --- top/bottom-8 + heads ---
    topk_kernel<<<1, 1024, 0, stream>>>(A_out, idx, N);
    finalize_kernel<<<1, 1, 0, stream>>>(pooled, Wcls, bcls, x3_out, Wcell, bcell,
                                         label, idx, out, N);
}